// SystemReadoutLayer_4930622456347
// MI455X (gfx1250) — compile-verified
//
#include <hip/hip_runtime.h>
#include <hip/hip_bf16.h>
#include <math.h>

typedef __attribute__((ext_vector_type(2))) float v2f;
typedef __attribute__((ext_vector_type(8))) float v8f;

#define NGRAPH 4096
#define DL 10
#define LRELU(x) ((x) > 0.f ? (x) : 0.01f * (x))

// ---------------- K0: per-graph partial  pg[b][j] = relu(gf[b]) . W_l1[j, 0:256]
__global__ __launch_bounds__(320) void pg_kernel(const float* __restrict__ gf,
                                                 const float* __restrict__ Wl1,
                                                 float* __restrict__ pg) {
  __shared__ float sgf[256];
  int b = blockIdx.x, tid = threadIdx.x;
  if (tid < 256) { float v = gf[b * 256 + tid]; sgf[tid] = v > 0.f ? v : 0.f; }
  __syncthreads();
  int w = tid >> 5, lane = tid & 31;   // w = 0..9 -> output j
  const float* wr = Wl1 + w * 512;     // first 256 cols apply to graph part
  float acc = 0.f;
  for (int k = lane; k < 256; k += 32) acc += sgf[k] * wr[k];
#pragma unroll
  for (int off = 16; off; off >>= 1) acc += __shfl_xor(acc, off, 32);
  if (lane == 0) pg[b * 16 + w] = acc;
}

// ---------------- K1: segment ranges from sorted ids (binary search)
__global__ void bounds_kernel(const int* __restrict__ seg, int* __restrict__ start, int nNodes) {
  int b = blockIdx.x * blockDim.x + threadIdx.x;
  if (b > NGRAPH) return;
  int lo = 0, hi = nNodes;
  while (lo < hi) { int mid = (lo + hi) >> 1; if (seg[mid] < b) lo = mid + 1; else hi = mid; }
  start[b] = lo;
}

// ---------------- K2: fused attention logits + segment softmax + weighted node sum
__global__ __launch_bounds__(256) void attn_kernel(
    const float* __restrict__ nf, const float* __restrict__ Wl1,
    const float* __restrict__ bl1, const float* __restrict__ Wl2,
    const float* __restrict__ bl2, const float* __restrict__ pg,
    const int* __restrict__ start, float* __restrict__ logits_ws,
    float* __restrict__ s_ws, float* __restrict__ sa_ws) {
  __shared__ float w1n[DL * 256];
  __shared__ float exch[256];
  __shared__ float red[256];
  __shared__ float pre_s[DL];
  __shared__ float w2_s[DL];
  __shared__ float misc[2];
  int b = blockIdx.x, tid = threadIdx.x;

  for (int idx = tid; idx < DL * 256; idx += 256) {
    int j = idx >> 8, k = idx & 255;
    w1n[idx] = Wl1[j * 512 + 256 + k];   // node-feature half of W_l1
  }
  if (tid < DL) { pre_s[tid] = pg[b * 16 + tid] + bl1[tid]; w2_s[tid] = Wl2[tid]; }
  if (tid == DL) misc[0] = bl2[0];
  __syncthreads();

  float pre_r[DL], w2_r[DL];
#pragma unroll
  for (int j = 0; j < DL; ++j) { pre_r[j] = pre_s[j]; w2_r[j] = w2_s[j]; }
  float b2 = misc[0];

  int n0 = start[b], n1 = start[b + 1];
  int wid = tid >> 5, lane = tid & 31;

  // ---- phase A: logits + running max (wave per node)
  float lmax = -3.402823466e38f;
  for (int n = n0 + wid; n < n1; n += 8) {
    const float* xr = nf + (size_t)n * 256;
    __builtin_prefetch(xr + 8 * 256, 0, 1);
    float acc[DL];
#pragma unroll
    for (int j = 0; j < DL; ++j) acc[j] = 0.f;
#pragma unroll
    for (int ki = 0; ki < 8; ++ki) {
      int k = lane + 32 * ki;
      float xv = xr[k];
#pragma unroll
      for (int j = 0; j < DL; ++j) acc[j] += xv * w1n[j * 256 + k];
    }
#pragma unroll
    for (int off = 16; off; off >>= 1) {
#pragma unroll
      for (int j = 0; j < DL; ++j) acc[j] += __shfl_xor(acc[j], off, 32);
    }
    float lg = b2;
#pragma unroll
    for (int j = 0; j < DL; ++j) { float h = pre_r[j] + acc[j]; h = LRELU(h); lg += w2_r[j] * h; }
    lg = LRELU(lg);
    if (lane == 0) logits_ws[n] = lg;
    lmax = fmaxf(lmax, lg);
  }
  if (lane == 0) red[wid] = lmax;
  __threadfence_block();   // make logits_ws writes visible within the block
  __syncthreads();
  if (tid == 0) {
    float m = red[0];
    for (int w = 1; w < 8; ++w) m = fmaxf(m, red[w]);
    misc[1] = m;
  }
  __syncthreads();
  float M = misc[1];

  // ---- phase B: exp, denominator, alpha-weighted feature sum (thread = feature)
  float sacc = 0.f, dacc = 0.f;
  for (int base = n0; base < n1; base += 256) {
    int cnt = min(256, n1 - base);
    __syncthreads();
    exch[tid] = (tid < cnt) ? __expf(logits_ws[base + tid] - M) : 0.f;
    __syncthreads();
    dacc += exch[tid];
    for (int i = 0; i < cnt; ++i) {
      if (base + i + 2 < n1) __builtin_prefetch(nf + (size_t)(base + i + 2) * 256 + tid, 0, 1);
      sacc += exch[i] * nf[(size_t)(base + i) * 256 + tid];
    }
  }
  __syncthreads();
  red[tid] = dacc;
  __syncthreads();
#pragma unroll
  for (int s = 128; s; s >>= 1) { if (tid < s) red[tid] += red[tid + s]; __syncthreads(); }
  float D = red[0];
  float denom = fmaxf(D, 1e-30f);
  s_ws[b * 256 + tid] = sacc / denom;        // weighted node sum (alpha-normalized)
  if (tid == 0) sa_ws[b] = D / denom;        // sum of alphas (1, or 0 for empty segment)
}

// ---------------- K3/K4: Out[m][n] = sum_k A[m][k]*W[n][k] + rowScale(m)*bias[n]
// fp32 WMMA 16x16x4, one wave per 16x16 output tile, K = 256.
__global__ __launch_bounds__(32) void gemm_nt_wmma(
    const float* __restrict__ A, const float* __restrict__ W,
    const float* __restrict__ bias, const float* __restrict__ rowScale,
    float* __restrict__ Out, int Ncols) {
  int m0 = blockIdx.x << 4, n0 = blockIdx.y << 4;
  int l = threadIdx.x;
  int hi = l >> 4, lm = l & 15;
  // A layout: lane l holds A[m0 + l%16][ks + 2*(l/16) + v], v = vgpr 0/1
  const float* arow = A + (size_t)(m0 + lm) * 256 + 2 * hi;
  // B layout: lane l holds B[K = ks + 2*(l/16) + v][N = l%16] = W[n0+l%16][K]
  const float* brow = W + (size_t)(n0 + lm) * 256 + 2 * hi;
  v8f c = {0.f, 0.f, 0.f, 0.f, 0.f, 0.f, 0.f, 0.f};
#pragma unroll 4
  for (int ks = 0; ks < 256; ks += 4) {
    v2f a = *(const v2f*)(arow + ks);
    v2f bb = *(const v2f*)(brow + ks);
    c = __builtin_amdgcn_wmma_f32_16x16x4_f32(false, a, false, bb, (short)0, c,
                                              false, false);
  }
  float bv = bias[n0 + lm];
#pragma unroll
  for (int v = 0; v < 8; ++v) {
    int row = m0 + v + 8 * hi;           // C/D layout: VGPR v -> M=v (lanes 0-15), M=v+8 (16-31)
    float sc = rowScale ? rowScale[row] : 1.f;
    Out[(size_t)row * Ncols + (n0 + lm)] = c[v] + sc * bv;
  }
}

// ---------------- K5: GRU gates + output
__global__ __launch_bounds__(256) void gru_kernel(const float* __restrict__ gi,
                                                  const float* __restrict__ gh,
                                                  const float* __restrict__ gf,
                                                  float* __restrict__ out) {
  int idx = blockIdx.x * 256 + threadIdx.x;
  int b = idx >> 8, k = idx & 255;
  const float* gib = gi + (size_t)b * 768;
  const float* ghb = gh + (size_t)b * 768;
  float r = 1.f / (1.f + __expf(-(gib[k] + ghb[k])));
  float z = 1.f / (1.f + __expf(-(gib[256 + k] + ghb[256 + k])));
  float nn = tanhf(gib[512 + k] + r * ghb[512 + k]);
  out[idx] = (1.f - z) * nn + z * gf[idx];
}

extern "C" void kernel_launch(void* const* d_in, const int* in_sizes, int n_in,
                              void* d_out, int out_size, void* d_ws, size_t ws_size,
                              hipStream_t stream) {
  const float* nf   = (const float*)d_in[0];
  const float* gf   = (const float*)d_in[1];
  const int*   seg  = (const int*)d_in[2];
  const float* Wmsg = (const float*)d_in[3];
  const float* bmsg = (const float*)d_in[4];
  const float* Wl1  = (const float*)d_in[5];
  const float* bl1  = (const float*)d_in[6];
  const float* Wl2  = (const float*)d_in[7];
  const float* bl2  = (const float*)d_in[8];
  const float* Wih  = (const float*)d_in[9];
  const float* Whh  = (const float*)d_in[10];
  const float* bih  = (const float*)d_in[11];
  const float* bhh  = (const float*)d_in[12];
  float* out = (float*)d_out;
  const int nNodes = in_sizes[2];

  float* ws = (float*)d_ws;
  size_t off = 0;
  auto alloc = [&](size_t n) { float* p = ws + off; off += (n + 63) & ~(size_t)63; return p; };
  float* pg     = alloc((size_t)NGRAPH * 16);
  int*   start  = (int*)alloc(NGRAPH + 1);
  float* logits = alloc((size_t)nNodes);
  float* s      = alloc((size_t)NGRAPH * 256);
  float* sa     = alloc(NGRAPH);
  float* ctx    = alloc((size_t)NGRAPH * 256);
  float* gi     = alloc((size_t)NGRAPH * 768);
  float* gh     = alloc((size_t)NGRAPH * 768);
  (void)ws_size; (void)n_in; (void)out_size;

  hipLaunchKernelGGL(pg_kernel, dim3(NGRAPH), dim3(320), 0, stream, gf, Wl1, pg);
  hipLaunchKernelGGL(bounds_kernel, dim3((NGRAPH + 1 + 255) / 256), dim3(256), 0, stream,
                     seg, start, nNodes);
  hipLaunchKernelGGL(attn_kernel, dim3(NGRAPH), dim3(256), 0, stream,
                     nf, Wl1, bl1, Wl2, bl2, pg, start, logits, s, sa);
  // context = s @ W_msg^T + (sum alpha) * b_msg
  hipLaunchKernelGGL(gemm_nt_wmma, dim3(NGRAPH / 16, 256 / 16), dim3(32), 0, stream,
                     s, Wmsg, bmsg, sa, ctx, 256);
  // gi = context @ W_ih^T + b_ih ; gh = graph_feats @ W_hh^T + b_hh
  hipLaunchKernelGGL(gemm_nt_wmma, dim3(NGRAPH / 16, 768 / 16), dim3(32), 0, stream,
                     ctx, Wih, bih, (const float*)nullptr, gi, 768);
  hipLaunchKernelGGL(gemm_nt_wmma, dim3(NGRAPH / 16, 768 / 16), dim3(32), 0, stream,
                     gf, Whh, bhh, (const float*)nullptr, gh, 768);
  hipLaunchKernelGGL(gru_kernel, dim3(NGRAPH), dim3(256), 0, stream, gi, gh, gf, out);
}